// PersonalizedPagerank_32341103739240
// MI455X (gfx1250) — compile-verified
//
#include <hip/hip_runtime.h>
#include <math.h>

#define ALPHA_F 0.85f

// ---------------------------------------------------------------------------
// Setup kernels (run once per launch; cost amortized over 100 iterations)
// ---------------------------------------------------------------------------

__global__ __launch_bounds__(256)
void init_zero_kernel(float* __restrict__ rowsum, int* __restrict__ deg,
                      float* __restrict__ scal, int* __restrict__ doneFlag, int N) {
    int i = blockIdx.x * blockDim.x + threadIdx.x;
    if (i < N) { rowsum[i] = 0.f; deg[i] = 0; }
    if (i == 0) {
        deg[N] = 0;
        scal[0] = 0.f;   // psum
        scal[1] = 0.f;   // pn_scale = (1-alpha)/psum (set by scan kernel)
        scal[2] = 0.f;   // err accumulator
        *doneFlag = 0;
    }
}

__global__ __launch_bounds__(256)
void psum_kernel(const float* __restrict__ p, float* __restrict__ scal, int N) {
    int stride = gridDim.x * blockDim.x;
    float acc = 0.f;
    for (int i = blockIdx.x * blockDim.x + threadIdx.x; i < N; i += stride)
        acc += p[i];
    // wave32 butterfly reduce
    acc += __shfl_xor(acc, 16, 32);
    acc += __shfl_xor(acc,  8, 32);
    acc += __shfl_xor(acc,  4, 32);
    acc += __shfl_xor(acc,  2, 32);
    acc += __shfl_xor(acc,  1, 32);
    __shared__ float lds[8];
    int lane = threadIdx.x & 31, wid = threadIdx.x >> 5;
    if (lane == 0) lds[wid] = acc;
    __syncthreads();
    if (threadIdx.x == 0) {
        float s = 0.f;
        #pragma unroll
        for (int i = 0; i < 8; ++i) s += lds[i];
        atomicAdd(&scal[0], s);
    }
}

__global__ __launch_bounds__(256)
void deg_rowsum_kernel(const int* __restrict__ src, const int* __restrict__ dst,
                       const float* __restrict__ weight,
                       int* __restrict__ deg, float* __restrict__ rowsum, int E) {
    int e = blockIdx.x * blockDim.x + threadIdx.x;
    if (e < E) {
        atomicAdd(&deg[dst[e]], 1);                 // in-degree histogram (CSR by dst)
        atomicAdd(&rowsum[src[e]], weight[e]);      // out-weight sums for normalization
    }
}

// One block, 1024 threads.  Exclusive scan of 4-aligned degrees -> rowstart
// (in place over deg), copy to scatter cursor, write (src=0, w=0) padding
// into the tail of every row so the SpMV can use b128 vector loads, compute
// inv(rowsum) in place, derive pn_scale.
__global__ __launch_bounds__(1024)
void scan_kernel(int* __restrict__ deg /*becomes aligned rowstart*/,
                 int* __restrict__ cursor,
                 float* __restrict__ rowsum /*becomes inv*/,
                 int* __restrict__ srcS, float* __restrict__ wS,
                 float* __restrict__ scal, int N) {
    const int tid = threadIdx.x;
    const int chunk = (N + 1023) / 1024;
    const int start = tid * chunk;
    const int end   = min(start + chunk, N);

    int s = 0;
    for (int i = start; i < end; ++i) s += (deg[i] + 3) & ~3;   // aligned degree

    __shared__ int lds[1024];
    lds[tid] = s;
    __syncthreads();
    // Hillis-Steele inclusive scan over 1024 thread-sums
    for (int off = 1; off < 1024; off <<= 1) {
        int v = (tid >= off) ? lds[tid - off] : 0;
        __syncthreads();
        lds[tid] += v;
        __syncthreads();
    }
    const int incl = lds[tid];
    int run = incl - s;   // exclusive prefix for this thread's chunk

    for (int i = start; i < end; ++i) {
        int t  = deg[i];
        int ta = (t + 3) & ~3;
        deg[i]    = run;          // aligned rowstart[i]
        cursor[i] = run;          // scatter cursor starts at aligned start
        // Fill padding slots with a zero-contribution edge (x[0] * 0.0f)
        for (int k = run + t; k < run + ta; ++k) { srcS[k] = 0; wS[k] = 0.f; }
        float r = rowsum[i];
        rowsum[i] = (r > 0.f) ? (1.f / r) : 0.f;   // inv in place
        run += ta;
    }
    if (tid == 1023) deg[N] = lds[1023];           // rowstart[N] = padded E'
    if (tid == 0)    scal[1] = (1.f - ALPHA_F) / scal[0];   // (1-alpha)/psum
}

__global__ __launch_bounds__(256)
void scatter_kernel(const int* __restrict__ src, const int* __restrict__ dst,
                    const float* __restrict__ weight, const float* __restrict__ inv,
                    int* __restrict__ cursor,
                    int* __restrict__ srcS, float* __restrict__ wS, int E) {
    int e = blockIdx.x * blockDim.x + threadIdx.x;
    if (e < E) {
        int s = src[e];
        int d = dst[e];
        float wn = weight[e] * inv[s];              // row-normalized edge weight
        int pos = atomicAdd(&cursor[d], 1);
        srcS[pos] = s;
        wS[pos]   = wn;
    }
}

__global__ __launch_bounds__(256)
void xinit_kernel(float* __restrict__ x, int N) {
    int i = blockIdx.x * blockDim.x + threadIdx.x;
    if (i < N) x[i] = 1.f / (float)N;
}

// ---------------------------------------------------------------------------
// Main iteration: fused pull-SpMV + AXPBY + L1-error reduction (no atomics on
// the hot path).  One wave32 per destination node; wave-stride over nodes.
// Rows are padded to multiples of 4 edges -> each lane streams (src,w) with
// global_load_b128 and issues 4 independent random gathers of x.
// ---------------------------------------------------------------------------
__global__ __launch_bounds__(256)
void spmv_kernel(const int* __restrict__ rowstart,
                 const int* __restrict__ srcS,
                 const float* __restrict__ wS,
                 const float* __restrict__ xo,
                 float* __restrict__ xn,
                 const float* __restrict__ p,
                 float* __restrict__ scal,
                 const int* __restrict__ doneFlag,
                 int N) {
    const int tid = blockIdx.x * blockDim.x + threadIdx.x;
    const int donev = *doneFlag;
    if (donev) {
        // Frozen: just propagate x through the ping-pong buffers.
        const int stride = gridDim.x * blockDim.x;
        for (int i = tid; i < N; i += stride) xn[i] = xo[i];
        return;
    }
    const float pscale = scal[1];
    const int lane = threadIdx.x & 31;
    const int wid  = threadIdx.x >> 5;
    const int gwid = tid >> 5;
    const int nw   = (gridDim.x * blockDim.x) >> 5;

    float errAcc = 0.f;
    for (int j = gwid; j < N; j += nw) {
        const int s0 = rowstart[j];
        const int s1 = rowstart[j + 1];      // aligned end (multiple of 4)
        // Warm the next node's edge block for this wave: the stride-nw row
        // jump defeats hardware stream prefetch; gfx1250 global_prefetch_b8.
        const int jn = j + nw;
        if (jn < N) {
            const int ns = rowstart[jn];
            __builtin_prefetch(&srcS[ns], 0, 3);
            __builtin_prefetch(&wS[ns], 0, 3);
        }
        float acc = 0.f;
        for (int e = s0 + (lane << 2); e < s1; e += 128) {
            const int4   si = *(const int4*)  &srcS[e];   // global_load_b128
            const float4 wi = *(const float4*)&wS[e];     // global_load_b128
            acc += xo[si.x] * wi.x;
            acc += xo[si.y] * wi.y;
            acc += xo[si.z] * wi.z;
            acc += xo[si.w] * wi.w;
        }
        // wave32 butterfly: every lane gets the node sum
        acc += __shfl_xor(acc, 16, 32);
        acc += __shfl_xor(acc,  8, 32);
        acc += __shfl_xor(acc,  4, 32);
        acc += __shfl_xor(acc,  2, 32);
        acc += __shfl_xor(acc,  1, 32);
        if (lane == 0) {
            float xnew = ALPHA_F * acc + pscale * p[j];
            errAcc += fabsf(xnew - xo[j]);
            xn[j] = xnew;
        }
    }
    __shared__ float lds[8];
    if (lane == 0) lds[wid] = errAcc;
    __syncthreads();
    if (threadIdx.x == 0) {
        float s = 0.f;
        #pragma unroll
        for (int i = 0; i < 8; ++i) s += lds[i];
        atomicAdd(&scal[2], s);   // one atomic per block per iteration
    }
}

__global__ void check_done_kernel(float* __restrict__ scal, int* __restrict__ doneFlag,
                                  float thresh) {
    if (scal[2] < thresh) *doneFlag = 1;   // done |= (err < thresh)
    scal[2] = 0.f;
}

// ---------------------------------------------------------------------------

extern "C" void kernel_launch(void* const* d_in, const int* in_sizes, int n_in,
                              void* d_out, int out_size, void* d_ws, size_t ws_size,
                              hipStream_t stream) {
    const int*   src    = (const int*)d_in[0];
    const int*   dst    = (const int*)d_in[1];
    const float* weight = (const float*)d_in[2];
    const float* p      = (const float*)d_in[3];
    const int E = in_sizes[0];
    const int N = in_sizes[3];
    const int Epad = E + 3 * N + 64;     // padded-CSR capacity

    // Workspace carve-up (256B aligned slices)
    char* ws = (char*)d_ws;
    size_t off = 0;
    auto carve = [&](size_t bytes) -> void* {
        void* ptr = ws + off;
        off = (off + bytes + 255) & ~(size_t)255;
        return ptr;
    };
    float* rowsum_inv = (float*)carve((size_t)N * 4);        // rowsum -> inv
    int*   rowstart   = (int*)  carve((size_t)(N + 1) * 4);  // deg -> aligned rowstart
    int*   cursor     = (int*)  carve((size_t)N * 4);
    float* xB         = (float*)carve((size_t)N * 4);        // ping-pong buffer
    int*   srcS       = (int*)  carve((size_t)Epad * 4);     // dst-sorted src (padded)
    float* wS         = (float*)carve((size_t)Epad * 4);     // dst-sorted weight (padded)
    float* scal       = (float*)carve(64);                   // psum, pn_scale, err
    int*   doneFlag   = (int*)  carve(64);

    float* xA = (float*)d_out;                               // final result lives here
    const float thresh = (float)N * 1e-6f;

    const int NB_N = (N + 255) / 256;
    const int NB_E = (E + 255) / 256;

    // ---- setup (one-time per launch) ----
    init_zero_kernel<<<NB_N, 256, 0, stream>>>(rowsum_inv, rowstart, scal, doneFlag, N);
    psum_kernel<<<128, 256, 0, stream>>>(p, scal, N);
    deg_rowsum_kernel<<<NB_E, 256, 0, stream>>>(src, dst, weight, rowstart, rowsum_inv, E);
    scan_kernel<<<1, 1024, 0, stream>>>(rowstart, cursor, rowsum_inv, srcS, wS, scal, N);
    scatter_kernel<<<NB_E, 256, 0, stream>>>(src, dst, weight, rowsum_inv, cursor,
                                             srcS, wS, E);
    xinit_kernel<<<NB_N, 256, 0, stream>>>(xA, N);

    // ---- 100 power iterations (frozen-after-convergence, on device) ----
    float* xo = xA;
    float* xn = xB;
    for (int k = 0; k < 100; ++k) {
        spmv_kernel<<<1024, 256, 0, stream>>>(rowstart, srcS, wS, xo, xn, p,
                                              scal, doneFlag, N);
        check_done_kernel<<<1, 1, 0, stream>>>(scal, doneFlag, thresh);
        float* t = xo; xo = xn; xn = t;
    }
    // k=99 (odd) wrote into xA == d_out; xo == d_out here.
}